// CountMeanOfFeatureInCluster_17093969838093
// MI455X (gfx1250) — compile-verified
//
#include <hip/hip_runtime.h>
#include <stdint.h>

// CountMeanOfFeatureInCluster on MI455X (gfx1250, wave32).
// Memory-bound: stream x (64MB) exactly once via async DMA-to-LDS, double
// buffered; distances via f32 WMMA GEMM (v_wmma_f32_16x16x4_f32).

#define EPSV 1e-6f
#define FDIM 2048
#define CNUM 7
#define BROWS 8192
#define ROWS_PER_BLOCK 32   // 2 waves x 16-row WMMA tiles
#define NTHREADS 64
#define KC 128              // K chunk staged in LDS
#define XSTRIDE (KC + 4)    // 132 floats: bank-conflict-free b64 fragment reads
#define BSTRIDE (KC + 4)

typedef float v2f __attribute__((ext_vector_type(2)));
typedef float v8f __attribute__((ext_vector_type(8)));

// ---- CDNA5 async DMA global->LDS (ASYNCcnt-tracked), inline asm ----
__device__ __forceinline__ void async_copy_b128(uint32_t lds_addr, const float* gptr) {
    asm volatile("global_load_async_to_lds_b128 %0, %1, off"
                 :: "v"(lds_addr), "v"((uint64_t)(uintptr_t)gptr)
                 : "memory");
}
#define WAIT_ASYNC(n) asm volatile("s_wait_asynccnt " #n ::: "memory")

// ws layout (floats): [0..7]=counts, [8..15]=elem_sums, [16..23]=h[c]=||m'||^2
__global__ void init_ws_kernel(float* __restrict__ ws) {
    if (threadIdx.x < 16) ws[threadIdx.x] = 0.0f;
}

// h[c] = sum_k (rm[c,k] - eps)^2
__global__ void __launch_bounds__(256)
hnorm_kernel(const float* __restrict__ rm, float* __restrict__ ws) {
    __shared__ float red[256];
    const int tid = threadIdx.x;
    for (int c = 0; c < CNUM; ++c) {
        float s = 0.0f;
        for (int k = tid; k < FDIM; k += 256) {
            float v = rm[c * FDIM + k] - EPSV;
            s += v * v;
        }
        red[tid] = s;
        __syncthreads();
        for (int off = 128; off > 0; off >>= 1) {
            if (tid < off) red[tid] += red[tid + off];
            __syncthreads();
        }
        if (tid == 0) ws[16 + c] = red[0];
        __syncthreads();
    }
}

// Issue one K-chunk of x (32 rows x 128 cols) as 16 async b128 DMAs per thread.
__device__ __forceinline__ void issue_x_chunk(const float* __restrict__ x,
                                              int rowBase, int kc,
                                              float* buf, int tid) {
    #pragma unroll 4
    for (int j = 0; j < 16; ++j) {
        const int linear = tid + j * NTHREADS;   // 0..1023 float4 slots
        const int row = linear >> 5;             // 32 float4 per row
        const int c4  = linear & 31;
        const float* g = x + (rowBase + row) * FDIM + kc + c4 * 4;
        const uint32_t l = (uint32_t)(uintptr_t)(buf + row * XSTRIDE + c4 * 4);
        async_copy_b128(l, g);
    }
}

// Main pass: per-row nearest cluster + per-cluster {count, row-sum} accumulation.
// score[b][c] = h[c] - 2 * (x_b . m'_c); WMMA B column 7 is all-ones -> row sum.
__global__ void __launch_bounds__(NTHREADS)
assign_kernel(const float* __restrict__ x, const float* __restrict__ rm,
              float* __restrict__ ws) {
    __shared__ __align__(16) float lds_x[2][ROWS_PER_BLOCK * XSTRIDE];
    __shared__ __align__(16) float lds_b[16 * BSTRIDE];
    __shared__ float lds_cnt[8];
    __shared__ float lds_sum[8];

    const int tid  = threadIdx.x;
    const int lane = tid & 31;
    const int wave = tid >> 5;          // 0..1
    const int col  = lane & 15;         // N (B/C/D) and M (A) index
    const int koff = (lane >> 4) * 2;   // K offset: lanes 0-15 -> K{0,1}, 16-31 -> K{2,3}

    if (tid < 8) { lds_cnt[tid] = 0.0f; lds_sum[tid] = 0.0f; }

    const int rowBase = blockIdx.x * ROWS_PER_BLOCK;
    const int waveRow = wave * 16;

    const float hval = (col < CNUM) ? ws[16 + col] : 0.0f;

    v8f acc = {};

    // prologue: kick off chunk 0 into buffer 0
    issue_x_chunk(x, rowBase, 0, lds_x[0], tid);
    int cur = 0;

    for (int kc = 0; kc < FDIM; kc += KC) {
        const bool more = (kc + KC) < FDIM;
        // overlap: issue next chunk's DMA into the other buffer
        if (more) issue_x_chunk(x, rowBase, kc + KC, lds_x[cur ^ 1], tid);

        // --- stage B chunk: 16 logical columns x KC, column-major in LDS ---
        {
            const int n  = tid >> 2;          // 0..15
            const int ks = (tid & 3) * 32;    // 0,32,64,96
            if (n < CNUM) {
                const float* src = rm + n * FDIM + kc + ks;
                #pragma unroll 8
                for (int i = 0; i < 32; ++i)
                    lds_b[n * BSTRIDE + ks + i] = src[i] - EPSV;
            } else {
                const float v = (n == CNUM) ? 1.0f : 0.0f;  // ones column -> row sums
                #pragma unroll 8
                for (int i = 0; i < 32; ++i)
                    lds_b[n * BSTRIDE + ks + i] = v;
            }
        }

        // current chunk's 16 DMAs done; next chunk's 16 may remain in flight
        if (more) { WAIT_ASYNC(16); } else { WAIT_ASYNC(0); }
        __syncthreads();

        // --- WMMA over the chunk: 32 x v_wmma_f32_16x16x4_f32 ---
        const float* xs = lds_x[cur];
        #pragma unroll 8
        for (int ks = 0; ks < KC; ks += 4) {
            v2f a = *reinterpret_cast<const v2f*>(
                &xs[(waveRow + col) * XSTRIDE + ks + koff]);
            v2f b = *reinterpret_cast<const v2f*>(
                &lds_b[col * BSTRIDE + ks + koff]);
            acc = __builtin_amdgcn_wmma_f32_16x16x4_f32(
                false, a, false, b, (short)0, acc, false, false);
        }
        __syncthreads();
        cur ^= 1;
    }

    // --- per-row argmin over 7 clusters + accumulation ---
    // C/D layout: VGPR r holds row M=r (lanes 0-15) / M=r+8 (lanes 16-31),
    // column = lane & 15. Reduce within each 16-lane half.
    const float BIG = 3.0e38f;
    #pragma unroll
    for (int r = 0; r < 8; ++r) {
        const float raw    = acc[r];
        const float rowsum = __shfl(raw, 7, 16);          // ones-column value
        float bv = (col < CNUM) ? (hval - 2.0f * raw) : BIG;
        int   bi = col;
        #pragma unroll
        for (int m = 1; m < 16; m <<= 1) {
            const float ov = __shfl_xor(bv, m, 16);
            const int   oi = __shfl_xor(bi, m, 16);
            if (ov < bv || (ov == bv && oi < bi)) { bv = ov; bi = oi; }
        }
        if (col == 0) {                 // one lane per row (lanes 0 and 16)
            atomicAdd(&lds_cnt[bi], 1.0f);
            atomicAdd(&lds_sum[bi], rowsum);
        }
    }
    __syncthreads();
    if (tid < CNUM) {
        atomicAdd(&ws[tid],     lds_cnt[tid]);
        atomicAdd(&ws[8 + tid], lds_sum[tid]);
    }
}

// out = (counts > 32) ? 0.1*scalar_mean + 0.9*rm : rm
__global__ void __launch_bounds__(256)
finalize_kernel(const float* __restrict__ rm, const float* __restrict__ ws,
                float* __restrict__ out) {
    const int i = blockIdx.x * blockDim.x + threadIdx.x;
    if (i >= CNUM * FDIM) return;
    const int c = i >> 11;                      // FDIM == 2048
    const float cnt = ws[c];
    const float sum = ws[8 + c];
    const float old = rm[i];
    const float sm  = sum / fmaxf(cnt * (float)FDIM, 1.0f);
    out[i] = (cnt > 32.0f) ? (0.1f * sm + 0.9f * old) : old;
}

extern "C" void kernel_launch(void* const* d_in, const int* in_sizes, int n_in,
                              void* d_out, int out_size, void* d_ws, size_t ws_size,
                              hipStream_t stream) {
    const float* x  = (const float*)d_in[0];   // [8192, 2048] f32
    const float* rm = (const float*)d_in[1];   // [7, 2048] f32
    float* out = (float*)d_out;                // [7, 2048] f32
    float* ws  = (float*)d_ws;

    init_ws_kernel<<<1, 32, 0, stream>>>(ws);
    hnorm_kernel<<<1, 256, 0, stream>>>(rm, ws);
    assign_kernel<<<BROWS / ROWS_PER_BLOCK, NTHREADS, 0, stream>>>(x, rm, ws);
    finalize_kernel<<<(CNUM * FDIM + 255) / 256, 256, 0, stream>>>(rm, ws, out);
}